// SOPWrapper_62612033241253
// MI455X (gfx1250) — compile-verified
//
#include <hip/hip_runtime.h>

// Second-order pooling (SOP): max-pooled outer products -> Newton-Schulz
// matrix sqrt (WMMA f32 16x16x4) -> fused L2 normalize.
// B=16, N=4096, C=64.

#define SOP_B 16
#define SOP_N 4096
#define SOP_C 64
#define SOP_CC (SOP_C * SOP_C)      // 4096 elements per batch matrix
#define POOL_CHUNK 256               // points per pooling block
#define TILE_PTS 64                  // points staged in LDS per step
#define NS_ITERS 8                   // Newton-Schulz iterations

typedef __attribute__((ext_vector_type(2))) float v2f;
typedef __attribute__((ext_vector_type(8))) float v8f;

// ---------------------------------------------------------------------------
// Kernel 0: init pooled accumulator to -inf (ws is poisoned by the harness).
// ---------------------------------------------------------------------------
__global__ __launch_bounds__(256) void sop_init_pooled(float* __restrict__ pooled) {
    const int i = blockIdx.x * 256 + threadIdx.x;
    if (i < SOP_B * SOP_CC) pooled[i] = -__builtin_inff();
}

// ---------------------------------------------------------------------------
// Kernel 1: pooled[b,i,j] = max_n x[b,n,i]*x[b,n,j]
// grid = (16 chunks, 16 batches), 256 threads; each thread owns a 4x4 tile of
// the 64x64 output; 256-point chunks merged via float atomicMax.
// Pure VALU work (max does not decompose into matmul), LDS-staged points.
// ---------------------------------------------------------------------------
__global__ __launch_bounds__(256) void sop_maxpool_outer(
        const float* __restrict__ x, float* __restrict__ pooled) {
    const int chunk = blockIdx.x;
    const int b     = blockIdx.y;
    const int tid   = threadIdx.x;

    __shared__ float xs[TILE_PTS * SOP_C];   // 16 KB

    const int i0 = (tid >> 4) << 2;          // row of 4x4 tile (0..60)
    const int j0 = (tid & 15) << 2;          // col of 4x4 tile (0..60)

    float acc[4][4];
#pragma unroll
    for (int a = 0; a < 4; ++a)
#pragma unroll
        for (int c = 0; c < 4; ++c) acc[a][c] = -__builtin_inff();

    const float* xb = x + ((size_t)b * SOP_N + (size_t)chunk * POOL_CHUNK) * SOP_C;

    for (int t = 0; t < POOL_CHUNK; t += TILE_PTS) {
        // 64 pts * 64 ch = 1024 float4; 256 threads * 4 each
        const float4* src = (const float4*)(xb + (size_t)t * SOP_C);
#pragma unroll
        for (int r = 0; r < 4; ++r)
            ((float4*)xs)[tid + r * 256] = src[tid + r * 256];

        if (t + TILE_PTS < POOL_CHUNK)
            __builtin_prefetch(xb + (size_t)(t + TILE_PTS) * SOP_C + tid * 4, 0, 0);

        __syncthreads();

#pragma unroll 4
        for (int p = 0; p < TILE_PTS; ++p) {
            const float4 vi = *(const float4*)(xs + p * SOP_C + i0);
            const float4 vj = *(const float4*)(xs + p * SOP_C + j0);
            const float fi[4] = {vi.x, vi.y, vi.z, vi.w};
            const float fj[4] = {vj.x, vj.y, vj.z, vj.w};
#pragma unroll
            for (int a = 0; a < 4; ++a)
#pragma unroll
                for (int c = 0; c < 4; ++c)
                    acc[a][c] = fmaxf(acc[a][c], fi[a] * fj[c]);
        }
        __syncthreads();
    }

    float* pb = pooled + (size_t)b * SOP_CC;
#pragma unroll
    for (int a = 0; a < 4; ++a)
#pragma unroll
        for (int c = 0; c < 4; ++c)
            atomicMax(&pb[(i0 + a) * SOP_C + (j0 + c)], acc[a][c]);
}

// ---------------------------------------------------------------------------
// 64x64x64 f32 GEMM in LDS using V_WMMA_F32_16X16X4_F32.
// D = alpha*(A@B) + beta*I.  8 waves each compute 2 of the 16 16x16 tiles.
// Fragment layouts per CDNA5 ISA 7.12.2:
//   A 16x4 f32 : lanes 0-15 hold M=lane, K={0,1}; lanes 16-31 K={2,3}
//   B 4x16 f32 : mirrored (row K striped across lanes)
//   C/D 16x16  : VGPR r -> rows r (lanes 0-15) and r+8 (lanes 16-31)
// ---------------------------------------------------------------------------
__device__ __forceinline__ void gemm64(const float* __restrict__ A,
                                       const float* __restrict__ B,
                                       float* __restrict__ D,
                                       float alpha, float beta,
                                       int wave, int lane) {
    const int l    = lane & 15;
    const int half = lane >> 4;          // 0 or 1
    const int kh   = half << 1;          // K offset 0 or 2

    for (int t = wave; t < 16; t += 8) {
        const int mt = t >> 2;
        const int nt = t & 3;
        v8f c = {0.f, 0.f, 0.f, 0.f, 0.f, 0.f, 0.f, 0.f};
        const float* Ar = A + (mt * 16 + l) * SOP_C;   // row of A
        const float* Bc = B + nt * 16 + l;             // column walker of B
#pragma unroll
        for (int kk = 0; kk < 16; ++kk) {
            const int kb = (kk << 2) + kh;
            v2f af, bf;
            af.x = Ar[kb];
            af.y = Ar[kb + 1];
            bf.x = Bc[kb * SOP_C];
            bf.y = Bc[(kb + 1) * SOP_C];
            c = __builtin_amdgcn_wmma_f32_16x16x4_f32(
                    /*neg_a=*/false, af, /*neg_b=*/false, bf,
                    /*c_mod=*/(short)0, c, /*reuse_a=*/false, /*reuse_b=*/false);
        }
        const int row0 = mt * 16 + (half << 3);
        const int col  = nt * 16 + l;
#pragma unroll
        for (int r = 0; r < 8; ++r) {
            float v = alpha * c[r];
            const int row = row0 + r;
            if (row == col) v += beta;
            D[row * SOP_C + col] = v;
        }
    }
}

// ---------------------------------------------------------------------------
// Kernel 2: per-batch Newton-Schulz sqrt + fused L2 normalize.
//   Y0 = A/||A||_F, Z0 = I
//   T = 1.5I - 0.5*Z@Y ; Y <- Y@T ; Z <- T@Z        (NS_ITERS times)
//   out = Y / max(||Y||_F, eps)    (sqrt(||A||_F) scale cancels under L2)
// 4 rotating 16 KB LDS buffers (64 KB total).
// ---------------------------------------------------------------------------
__global__ __launch_bounds__(256) void sop_sqrtm_normalize(
        const float* __restrict__ pooled, float* __restrict__ out) {
    const int b    = blockIdx.x;
    const int tid  = threadIdx.x;
    const int wave = tid >> 5;
    const int lane = tid & 31;

    __shared__ float smem[4 * SOP_CC];   // 64 KB
    float* Y = smem;
    float* Z = smem + SOP_CC;
    float* T = smem + 2 * SOP_CC;
    float* S = smem + 3 * SOP_CC;

    const float* Ag = pooled + (size_t)b * SOP_CC;

    // Stage A into Y, compute Frobenius norm (reduce in T scratch)
    for (int k = tid; k < SOP_CC; k += 256) Y[k] = Ag[k];
    __syncthreads();
    float s = 0.f;
    for (int k = tid; k < SOP_CC; k += 256) s += Y[k] * Y[k];
    T[tid] = s;
    __syncthreads();
    for (int off = 128; off > 0; off >>= 1) {
        if (tid < off) T[tid] += T[tid + off];
        __syncthreads();
    }
    const float inv = 1.0f / fmaxf(sqrtf(T[0]), 1e-20f);
    __syncthreads();   // done reading T scratch

    for (int k = tid; k < SOP_CC; k += 256) {
        Y[k] *= inv;
        Z[k] = ((k >> 6) == (k & 63)) ? 1.0f : 0.0f;
    }
    __syncthreads();

    for (int it = 0; it < NS_ITERS; ++it) {
        gemm64(Z, Y, T, -0.5f, 1.5f, wave, lane);  // T = 1.5I - 0.5*Z@Y
        __syncthreads();
        gemm64(Y, T, S, 1.0f, 0.0f, wave, lane);   // S = Y@T   (Ynew)
        __syncthreads();
        gemm64(T, Z, Y, 1.0f, 0.0f, wave, lane);   // Yold <- T@Z (Znew)
        __syncthreads();
        float* oy = Y; float* oz = Z;
        Y = S;    // new Y
        Z = oy;   // new Z
        S = oz;   // spare
    }

    // Fused L2 normalization of flattened Y
    float s2 = 0.f;
    for (int k = tid; k < SOP_CC; k += 256) s2 += Y[k] * Y[k];
    T[tid] = s2;
    __syncthreads();
    for (int off = 128; off > 0; off >>= 1) {
        if (tid < off) T[tid] += T[tid + off];
        __syncthreads();
    }
    const float invn = 1.0f / fmaxf(sqrtf(T[0]), 1e-12f);
    float* ob = out + (size_t)b * SOP_CC;
    for (int k = tid; k < SOP_CC; k += 256) ob[k] = Y[k] * invn;
}

// ---------------------------------------------------------------------------
extern "C" void kernel_launch(void* const* d_in, const int* in_sizes, int n_in,
                              void* d_out, int out_size, void* d_ws, size_t ws_size,
                              hipStream_t stream) {
    const float* x   = (const float*)d_in[0];
    float*       out = (float*)d_out;
    float*       pooled = (float*)d_ws;   // 16*4096 f32 = 256 KB

    (void)in_sizes; (void)n_in; (void)out_size; (void)ws_size;

    sop_init_pooled<<<(SOP_B * SOP_CC + 255) / 256, 256, 0, stream>>>(pooled);
    sop_maxpool_outer<<<dim3(SOP_N / POOL_CHUNK, SOP_B), 256, 0, stream>>>(x, pooled);
    sop_sqrtm_normalize<<<SOP_B, 256, 0, stream>>>(pooled, out);
}